// VLAWithExpert_79431125172994
// MI455X (gfx1250) — compile-verified
//
#include <hip/hip_runtime.h>
#include <hip/hip_bf16.h>
#include <math.h>
#include <stdint.h>

// Problem constants (match reference)
#define Lh   4
#define Bh   16
#define Sh   16
#define KVh  2048
#define Hh   6
#define Dh   128
#define HIDh 768
#define FFh  2048
#define TKh  (KVh + Sh)          // 2064 total keys
#define Mrows (Bh * Sh)          // 256
#define SCALEh 0.08838834764831845f  // 1/sqrt(128)

typedef float v2f __attribute__((ext_vector_type(2)));
typedef float v8f __attribute__((ext_vector_type(8)));

// ---------------------------------------------------------------------------
// fp32 WMMA GEMM with double-buffered async-to-LDS staging.
//   C[M,N] (+)= A[M,K] @ B[K,N] (+ bias)
// Block = 128 threads (4 waves). Block tile: 32(M) x 128(N).
// Wave (mw = w&1, nw = w>>1) computes rows mw*16..+15, cols nw*64..+63
// as 4 accumulators of 16x16, K consumed in chunks of 16 via
// V_WMMA_F32_16X16X4_F32 (4 k-steps x 4 n-tiles = 16 wmma per chunk).
// LDS tiles are filled with GLOBAL_LOAD_ASYNC_TO_LDS_B128 (ASYNCcnt),
// double buffered: chunk c+1 streams in while chunk c is consumed.
// The LDS destination is passed as (unsigned)(uintptr_t)&smem[...]:
// low 32 bits of a flat LDS address ARE the LDS byte offset, and the
// ptrtoint capture forces the compiler to treat smem as asm-written.
// Requires M%32==0, N%128==0, K%16==0 (true for all calls here).
// ---------------------------------------------------------------------------
#define ASTRf 20                  // padded A row stride (floats)
#define ABUFf (32 * ASTRf)        // 640 floats per A buffer
#define BBASEf (2 * ABUFf)        // 1280: B region start
#define BSTRf 136                 // padded B row stride (floats)
#define BBUFf (16 * BSTRf)        // 2176 floats per B buffer
#define SMEMf (BBASEf + 2 * BBUFf)  // 5632 floats = 22528 B

__global__ void __launch_bounds__(128)
gemm_f32_wmma(const float* __restrict__ A, const float* __restrict__ Bm,
              const float* __restrict__ bias, float* __restrict__ C,
              int M, int N, int K, int accumulate)
{
    __shared__ float smem[SMEMf];

    const int t    = threadIdx.x;
    const int wave = t >> 5;
    const int lane = t & 31;
    const int mw   = wave & 1;
    const int nw   = wave >> 1;
    const int m0   = blockIdx.y * 32;
    const int n0   = blockIdx.x * 128;
    const int khalf = (lane >> 4) * 2;   // 0 or 2
    const int lm    = lane & 15;

    // staging geometry (per thread)
    const int aRow = t >> 2;          // 0..31
    const int aK   = (t & 3) * 4;     // 0,4,8,12
    const int bRow = t >> 3;          // 0..15
    const int bCol = (t & 7) * 16;    // 0..112

    v8f acc[4] = {{}, {}, {}, {}};

    const int nchunks = K >> 4;

    // prologue: stream chunk 0 into buffer 0
    {
        const float*   ga = A + (size_t)(m0 + aRow) * K + aK;
        const unsigned la = (unsigned)(uintptr_t)&smem[aRow * ASTRf + aK];
        asm volatile("global_load_async_to_lds_b128 %0, %1, off"
                     :: "v"(la), "v"(ga) : "memory");
#pragma unroll
        for (int i = 0; i < 4; ++i) {
            const float*   gb = Bm + (size_t)bRow * N + n0 + bCol + i * 4;
            const unsigned lb =
                (unsigned)(uintptr_t)&smem[BBASEf + bRow * BSTRf + bCol + i * 4];
            asm volatile("global_load_async_to_lds_b128 %0, %1, off"
                         :: "v"(lb), "v"(gb) : "memory");
        }
    }

    for (int c = 0; c < nchunks; ++c) {
        const int buf = c & 1;
        asm volatile("s_wait_asynccnt 0x0" ::: "memory");
        __syncthreads();

        if (c + 1 < nchunks) {         // stream next chunk into other buffer
            const int kc = (c + 1) << 4;
            const int nb = buf ^ 1;
            const float*   ga = A + (size_t)(m0 + aRow) * K + kc + aK;
            const unsigned la =
                (unsigned)(uintptr_t)&smem[nb * ABUFf + aRow * ASTRf + aK];
            asm volatile("global_load_async_to_lds_b128 %0, %1, off"
                         :: "v"(la), "v"(ga) : "memory");
#pragma unroll
            for (int i = 0; i < 4; ++i) {
                const float*   gb = Bm + (size_t)(kc + bRow) * N + n0 + bCol + i * 4;
                const unsigned lb = (unsigned)(uintptr_t)
                    &smem[BBASEf + nb * BBUFf + bRow * BSTRf + bCol + i * 4];
                asm volatile("global_load_async_to_lds_b128 %0, %1, off"
                             :: "v"(lb), "v"(gb) : "memory");
            }
        }

        const float* __restrict__ sa = smem + buf * ABUFf + (mw * 16 + lm) * ASTRf;
        const float* __restrict__ sb = smem + BBASEf + buf * BBUFf + nw * 64 + lm;
#pragma unroll
        for (int ks = 0; ks < 16; ks += 4) {
            v2f a;
            a[0] = sa[ks + khalf];
            a[1] = sa[ks + khalf + 1];
#pragma unroll
            for (int nt = 0; nt < 4; ++nt) {
                v2f b;
                b[0] = sb[(ks + khalf) * BSTRf + nt * 16];
                b[1] = sb[(ks + khalf + 1) * BSTRf + nt * 16];
                acc[nt] = __builtin_amdgcn_wmma_f32_16x16x4_f32(
                    /*neg_a=*/false, a, /*neg_b=*/false, b,
                    /*c_mod=*/(short)0, acc[nt],
                    /*reuse_a=*/false, /*reuse_b=*/false);
            }
        }
        __syncthreads();   // everyone done with buf before chunk c+2 overwrites it
    }

    const int rbase = (lane >> 4) * 8;
#pragma unroll
    for (int nt = 0; nt < 4; ++nt) {
        const int   ncol = n0 + nw * 64 + nt * 16 + lm;
        const float bvv  = bias ? bias[ncol] : 0.0f;
#pragma unroll
        for (int r = 0; r < 8; ++r) {
            const int    mrow = m0 + mw * 16 + rbase + r;
            const size_t idx  = (size_t)mrow * N + ncol;
            const float  val  = acc[nt][r] + bvv;
            if (accumulate) C[idx] += val;
            else            C[idx]  = val;
        }
    }
}

// ---------------------------------------------------------------------------
// RMSNorm: one block per row of HID=768
// ---------------------------------------------------------------------------
__global__ void __launch_bounds__(256)
rmsnorm_kernel(const float* __restrict__ x, const float* __restrict__ w,
               float* __restrict__ out)
{
    const int row = blockIdx.x;
    __shared__ float red[256];
    float s = 0.0f;
    for (int j = threadIdx.x; j < HIDh; j += 256) {
        const float v = x[(size_t)row * HIDh + j];
        s += v * v;
    }
    red[threadIdx.x] = s;
    __syncthreads();
    for (int st = 128; st > 0; st >>= 1) {
        if (threadIdx.x < st) red[threadIdx.x] += red[threadIdx.x + st];
        __syncthreads();
    }
    const float inv = rsqrtf(red[0] / (float)HIDh + 1e-6f);
    for (int j = threadIdx.x; j < HIDh; j += 256)
        out[(size_t)row * HIDh + j] = x[(size_t)row * HIDh + j] * inv * w[j];
}

// ---------------------------------------------------------------------------
// RoPE in-place on a (B*S, H*D) buffer. Block = 384 threads = H * D/2.
// ---------------------------------------------------------------------------
__global__ void __launch_bounds__(384)
rope_kernel(float* __restrict__ buf, const long long* __restrict__ pos)
{
    const int row = blockIdx.x;           // b*S + s
    const int h   = threadIdx.x / 64;
    const int i   = threadIdx.x % 64;
    const float p  = (float)pos[row];
    const float ts = powf(10000.0f, (float)i / 64.0f);
    const float r  = p / ts;
    const float sn = sinf(r), cs = cosf(r);
    const size_t i1 = (size_t)row * (Hh * Dh) + h * Dh + i;
    const size_t i2 = i1 + 64;
    const float x1 = buf[i1], x2 = buf[i2];
    buf[i1] = x1 * cs - x2 * sn;
    buf[i2] = x2 * cs + x1 * sn;
}

// ---------------------------------------------------------------------------
// Attention mask cumsum: cum[b][j] = sum_{t<=j} att[b][t]
// ---------------------------------------------------------------------------
__global__ void cumsum_kernel(const int* __restrict__ att, int* __restrict__ cum)
{
    const int b = threadIdx.x;
    if (b >= Bh) return;
    int s = 0;
    for (int j = 0; j < TKh; ++j) {
        s += att[(size_t)b * TKh + j];
        cum[(size_t)b * TKh + j] = s;
    }
}

// ---------------------------------------------------------------------------
// Attention scores: scores[bh][q][j] = q.k * scale + mask_bias
// grid = (B*H, ceil(TK/256)), block = 256 (one key per thread); Q tile in LDS.
// ---------------------------------------------------------------------------
__global__ void __launch_bounds__(256)
attn_scores_kernel(const float* __restrict__ qb, const float* __restrict__ kcache_l,
                   const float* __restrict__ kb, const int* __restrict__ cum,
                   const unsigned char* __restrict__ pad, float* __restrict__ scores)
{
    const int bh = blockIdx.x;
    const int b  = bh / Hh, h = bh % Hh;
    __shared__ float qs[Sh][Dh];
    for (int t = threadIdx.x; t < Sh * Dh; t += 256) {
        const int qq = t >> 7, d = t & 127;
        qs[qq][d] = qb[(size_t)(b * Sh + qq) * (Hh * Dh) + h * Dh + d];
    }
    __syncthreads();

    const int j = blockIdx.y * 256 + threadIdx.x;
    if (j >= TKh) return;

    const float* __restrict__ kv =
        (j < KVh) ? kcache_l + ((size_t)(b * Hh + h) * KVh + j) * Dh
                  : kb + (size_t)(b * Sh + (j - KVh)) * (Hh * Dh) + h * Dh;

    float acc[Sh];
#pragma unroll
    for (int qq = 0; qq < Sh; ++qq) acc[qq] = 0.0f;
    for (int d = 0; d < Dh; ++d) {
        const float kval = kv[d];
#pragma unroll
        for (int qq = 0; qq < Sh; ++qq) acc[qq] += kval * qs[qq][d];
    }

    const int  cj = cum[(size_t)b * TKh + j];
    const bool pj = pad[(size_t)b * TKh + j] != 0;
#pragma unroll
    for (int qq = 0; qq < Sh; ++qq) {
        const int  cq = cum[(size_t)b * TKh + KVh + qq];
        const bool pq = pad[(size_t)b * TKh + KVh + qq] != 0;
        const bool ok = (cj <= cq) && pj && pq;
        scores[(size_t)(bh * Sh + qq) * TKh + j] =
            acc[qq] * SCALEh + (ok ? 0.0f : -1.0e9f);
    }
}

// ---------------------------------------------------------------------------
// Row softmax over TK columns (in place). grid = B*H*S rows.
// ---------------------------------------------------------------------------
__global__ void __launch_bounds__(256)
softmax_kernel(float* __restrict__ scores)
{
    float* __restrict__ p = scores + (size_t)blockIdx.x * TKh;
    __shared__ float red[256];

    float m = -1.0e30f;
    for (int j = threadIdx.x; j < TKh; j += 256) m = fmaxf(m, p[j]);
    red[threadIdx.x] = m;
    __syncthreads();
    for (int st = 128; st > 0; st >>= 1) {
        if (threadIdx.x < st) red[threadIdx.x] = fmaxf(red[threadIdx.x], red[threadIdx.x + st]);
        __syncthreads();
    }
    m = red[0];
    __syncthreads();

    float s = 0.0f;
    for (int j = threadIdx.x; j < TKh; j += 256) {
        const float e = expf(p[j] - m);
        p[j] = e;
        s += e;
    }
    red[threadIdx.x] = s;
    __syncthreads();
    for (int st = 128; st > 0; st >>= 1) {
        if (threadIdx.x < st) red[threadIdx.x] += red[threadIdx.x + st];
        __syncthreads();
    }
    const float inv = 1.0f / red[0];
    for (int j = threadIdx.x; j < TKh; j += 256) p[j] *= inv;
}

// ---------------------------------------------------------------------------
// o[b][q][h][d] = sum_j attn[bh][q][j] * V[j][d]
// grid = B*H, block = 256: thread owns d = tid&127, query-group g = tid>>7.
// ---------------------------------------------------------------------------
__global__ void __launch_bounds__(256)
attn_av_kernel(const float* __restrict__ scores, const float* __restrict__ vcache_l,
               const float* __restrict__ vb, float* __restrict__ ob)
{
    const int bh = blockIdx.x;
    const int b  = bh / Hh, h = bh % Hh;
    const int d  = threadIdx.x & 127;
    const int g  = threadIdx.x >> 7;   // 0 or 1 -> queries g*8..g*8+7

    float acc[8];
#pragma unroll
    for (int qi = 0; qi < 8; ++qi) acc[qi] = 0.0f;

    for (int j = 0; j < TKh; ++j) {
        const float vval =
            (j < KVh) ? vcache_l[((size_t)(b * Hh + h) * KVh + j) * Dh + d]
                      : vb[(size_t)(b * Sh + (j - KVh)) * (Hh * Dh) + h * Dh + d];
#pragma unroll
        for (int qi = 0; qi < 8; ++qi)
            acc[qi] += scores[(size_t)(bh * Sh + g * 8 + qi) * TKh + j] * vval;
    }
#pragma unroll
    for (int qi = 0; qi < 8; ++qi)
        ob[(size_t)(b * Sh + g * 8 + qi) * (Hh * Dh) + h * Dh + d] = acc[qi];
}

// ---------------------------------------------------------------------------
// SwiGLU elementwise: out = silu(g) * u
// ---------------------------------------------------------------------------
__global__ void __launch_bounds__(256)
silu_mul_kernel(const float* __restrict__ g, const float* __restrict__ u,
                float* __restrict__ out, int n)
{
    const int i = blockIdx.x * 256 + threadIdx.x;
    if (i < n) {
        const float gv = g[i];
        out[i] = gv / (1.0f + expf(-gv)) * u[i];
    }
}

// ---------------------------------------------------------------------------
// Host orchestration
// ---------------------------------------------------------------------------
extern "C" void kernel_launch(void* const* d_in, const int* in_sizes, int n_in,
                              void* d_out, int out_size, void* d_ws, size_t ws_size,
                              hipStream_t stream)
{
    (void)in_sizes; (void)n_in; (void)out_size; (void)ws_size;

    const float*         inputs_embeds = (const float*)d_in[0];
    const long long*     position_ids  = (const long long*)d_in[1];
    const unsigned char* pad_masks     = (const unsigned char*)d_in[2];
    const int*           att_masks     = (const int*)d_in[3];
    const float*         k_cache       = (const float*)d_in[4];
    const float*         v_cache       = (const float*)d_in[5];
    const float*         Wq  = (const float*)d_in[6];
    const float*         bq  = (const float*)d_in[7];
    const float*         Wk  = (const float*)d_in[8];
    const float*         bk  = (const float*)d_in[9];
    const float*         Wv  = (const float*)d_in[10];
    const float*         bv  = (const float*)d_in[11];
    const float*         Wo  = (const float*)d_in[12];
    const float*         Wg  = (const float*)d_in[13];
    const float*         Wu  = (const float*)d_in[14];
    const float*         Wd  = (const float*)d_in[15];
    const float*         ln1 = (const float*)d_in[16];
    const float*         ln2 = (const float*)d_in[17];
    const float*         norm_f = (const float*)d_in[18];
    float* out = (float*)d_out;

    // Workspace layout (floats)
    float* ws = (float*)d_ws;
    const size_t nHid = (size_t)Mrows * HIDh;   // 196608
    const size_t nFf  = (size_t)Mrows * FFh;    // 524288
    float* h      = ws;            ws += nHid;
    float* hn     = ws;            ws += nHid;
    float* qb     = ws;            ws += nHid;
    float* kb     = ws;            ws += nHid;
    float* vb     = ws;            ws += nHid;
    float* ob     = ws;            ws += nHid;
    float* ffg    = ws;            ws += nFf;
    float* ffu    = ws;            ws += nFf;
    float* ffa    = ws;            ws += nFf;
    float* scores = ws;            ws += (size_t)Bh * Hh * Sh * TKh;  // ~3.17M
    int*   cum    = (int*)ws;

    // h = inputs_embeds (residual stream)
    hipMemcpyAsync(h, inputs_embeds, nHid * sizeof(float),
                   hipMemcpyDeviceToDevice, stream);

    // attention mask cumsum
    cumsum_kernel<<<1, 32, 0, stream>>>(att_masks, cum);

    const dim3 gemm768(HIDh / 128, Mrows / 32);   // N=768 : 6 x 8
    const dim3 gemmFF (FFh  / 128, Mrows / 32);   // N=2048: 16 x 8
    const dim3 scoreGrid(Bh * Hh, (TKh + 255) / 256);

    for (int l = 0; l < Lh; ++l) {
        const size_t wOff  = (size_t)l * HIDh * (Hh * Dh);
        const size_t bOff  = (size_t)l * (Hh * Dh);
        const size_t kvOff = (size_t)l * Bh * Hh * KVh * Dh;

        // hn = rmsnorm(h, ln1[l])
        rmsnorm_kernel<<<Mrows, 256, 0, stream>>>(h, ln1 + (size_t)l * HIDh, hn);

        // q,k,v projections (+bias)
        gemm_f32_wmma<<<gemm768, 128, 0, stream>>>(hn, Wq + wOff, bq + bOff, qb,
                                                   Mrows, Hh * Dh, HIDh, 0);
        gemm_f32_wmma<<<gemm768, 128, 0, stream>>>(hn, Wk + wOff, bk + bOff, kb,
                                                   Mrows, Hh * Dh, HIDh, 0);
        gemm_f32_wmma<<<gemm768, 128, 0, stream>>>(hn, Wv + wOff, bv + bOff, vb,
                                                   Mrows, Hh * Dh, HIDh, 0);

        // RoPE on q and k
        rope_kernel<<<Mrows, 384, 0, stream>>>(qb, position_ids);
        rope_kernel<<<Mrows, 384, 0, stream>>>(kb, position_ids);

        // attention
        attn_scores_kernel<<<scoreGrid, 256, 0, stream>>>(
            qb, k_cache + kvOff, kb, cum, pad_masks, scores);
        softmax_kernel<<<Bh * Hh * Sh, 256, 0, stream>>>(scores);
        attn_av_kernel<<<Bh * Hh, 256, 0, stream>>>(scores, v_cache + kvOff, vb, ob);

        // h += ob @ Wo[l]
        gemm_f32_wmma<<<gemm768, 128, 0, stream>>>(
            ob, Wo + (size_t)l * (Hh * Dh) * HIDh, nullptr, h,
            Mrows, HIDh, Hh * Dh, 1);

        // hn = rmsnorm(h, ln2[l])
        rmsnorm_kernel<<<Mrows, 256, 0, stream>>>(h, ln2 + (size_t)l * HIDh, hn);

        // FFN
        gemm_f32_wmma<<<gemmFF, 128, 0, stream>>>(
            hn, Wg + (size_t)l * HIDh * FFh, nullptr, ffg, Mrows, FFh, HIDh, 0);
        gemm_f32_wmma<<<gemmFF, 128, 0, stream>>>(
            hn, Wu + (size_t)l * HIDh * FFh, nullptr, ffu, Mrows, FFh, HIDh, 0);
        silu_mul_kernel<<<(int)((nFf + 255) / 256), 256, 0, stream>>>(
            ffg, ffu, ffa, (int)nFf);
        gemm_f32_wmma<<<gemm768, 128, 0, stream>>>(
            ffa, Wd + (size_t)l * FFh * HIDh, nullptr, h, Mrows, HIDh, FFh, 1);
    }

    // out = rmsnorm(h, norm_f)
    rmsnorm_kernel<<<Mrows, 256, 0, stream>>>(h, norm_f, out);
}